// GraphConvDecoder_65180423684863
// MI455X (gfx1250) — compile-verified
//
#include <hip/hip_runtime.h>
#include <hip/hip_bf16.h>
#include <math.h>

// ---------------------------------------------------------------------------
// GraphConvDecoder for MI455X (gfx1250).
// All GEMMs run through V_WMMA_F32_16X16X4_F32 with a guard-free inner loop:
//   - weights pre-padded to (Mp x Kp), Mp%16==0, Kp%4==0 (prep kernel)
//   - activations stored point-major (B, NPTS, Kp) with zeroed pad columns
//   => per k-step: one global_load_b64 (A), one global_load_b64 (B), one WMMA.
// ---------------------------------------------------------------------------

typedef __attribute__((ext_vector_type(2))) float v2f;
typedef __attribute__((ext_vector_type(8))) float v8f;

#define NPTS 4096
#define NB   4

__device__ __forceinline__ int clampi(int v, int lo, int hi) {
    return v < lo ? lo : (v > hi ? hi : v);
}

// ---------------------------------------------------------------------------
// Pad W (M x K, row-major) -> Wp (Mp x Kp) with zero fill.
// ---------------------------------------------------------------------------
__global__ void pad_w_kernel(const float* __restrict__ src, int M, int K,
                             float* __restrict__ dst, int Mp, int Kp)
{
    int t = blockIdx.x * 256 + threadIdx.x;
    int total = Mp * Kp;
    if (t >= total) return;
    int k = t % Kp;
    int m = t / Kp;
    dst[t] = (m < M && k < K) ? src[(size_t)m * K + k] : 0.0f;
}

// ---------------------------------------------------------------------------
// Dual-segment GEMM:  out[b,n,:] = act( W1*X1[b,n,:] + W2*X2[b,n,:] + bias )
//   W: (Mp x Kp) row-major (padded).  X: (B, NPTS, Kp) point-major (padded).
//   out: (B, NPTS, Mstride) point-major.
// One wave computes one 16x16 (M x N-points) tile via 16x16x4 fp32 WMMAs.
// Fragment layouts per CDNA5 ISA 7.12.2:
//   A 16x4 f32 : lane m=lane&15, VGPR0/1 = A[m][2*half+0/1]  -> b64 load
//   B 4x16 f32 : lane n=lane&15, VGPR0/1 = B[2*half+0/1][n]  -> b64 load
//   C/D 16x16  : VGPR r -> row (8*half + r), col lane&15
// ---------------------------------------------------------------------------
__global__ void gemm_wmma_kernel(
    const float* __restrict__ W1, const float* __restrict__ X1, int K1p,
    const float* __restrict__ W2, const float* __restrict__ X2, int K2p,
    const float* __restrict__ bias, float* __restrict__ out,
    int M, int Mstride, int do_relu, int total_tiles)
{
    int wave = threadIdx.x >> 5;
    int lane = threadIdx.x & 31;
    int tile = blockIdx.x * 8 + wave;
    if (tile >= total_tiles) return;            // wave-uniform

    const int tiles_n = NPTS / 16;
    int tiles_m = (M + 15) >> 4;
    int per_b   = tiles_m * tiles_n;
    int b  = tile / per_b;
    int r  = tile - b * per_b;
    int tm = r / tiles_n;
    int tn = r - tm * tiles_n;

    int half = lane >> 4;      // 0 or 1
    int lm   = lane & 15;
    int mrow = tm * 16 + lm;   // always < Mp (W padded)
    int ncol = tn * 16 + lm;

    v8f acc = {};
    for (int seg = 0; seg < 2; ++seg) {
        const float* W = seg ? W2 : W1;
        const float* X = seg ? X2 : X1;
        int Kp = seg ? K2p : K1p;
        if (Kp == 0) continue;
        const float* Wrow = W + (size_t)mrow * Kp + half * 2;
        const float* Xrow = X + ((size_t)b * NPTS + ncol) * Kp + half * 2;
#pragma unroll 2
        for (int k = 0; k < Kp; k += 4) {
            v2f a  = *(const v2f*)(Wrow + k);
            v2f bf = *(const v2f*)(Xrow + k);
            acc = __builtin_amdgcn_wmma_f32_16x16x4_f32(
                /*neg_a=*/false, a, /*neg_b=*/false, bf,
                /*c_mod=*/(short)0, acc, /*reuse_a=*/false, /*reuse_b=*/false);
        }
    }

    float* Op = out + ((size_t)b * NPTS + ncol) * Mstride;
#pragma unroll
    for (int rr = 0; rr < 8; ++rr) {
        int m = tm * 16 + half * 8 + rr;
        if (m < M) {
            float v = acc[rr] + bias[m];
            if (do_relu) v = fmaxf(v, 0.0f);
            Op[m] = v;
        }
    }
}

// ---------------------------------------------------------------------------
// X[b,n,c] = concat([loc(3); A(CA); Bf(CB)])[c], zero for c >= 3+CA+CB (pad).
// loc is channel-major (B,3,NPTS).  A/Bf layout per flag: 1 = point-major
// (B,NPTS,C), 0 = channel-major (B,C,NPTS).
// ---------------------------------------------------------------------------
__device__ __forceinline__ float read_src(const float* __restrict__ loc,
                                          const float* __restrict__ A, int CA, int A_pm,
                                          const float* __restrict__ Bf, int CB, int B_pm,
                                          int b, int c, int n)
{
    if (c < 3) return loc[((size_t)b * 3 + c) * NPTS + n];
    int ca = c - 3;
    if (ca < CA)
        return A_pm ? A[((size_t)b * NPTS + n) * CA + ca]
                    : A[((size_t)b * CA + ca) * NPTS + n];
    int cb = ca - CA;
    return B_pm ? Bf[((size_t)b * NPTS + n) * CB + cb]
                : Bf[((size_t)b * CB + cb) * NPTS + n];
}

__global__ void concat_pm_kernel(const float* __restrict__ loc,
                                 const float* __restrict__ A, int CA, int A_pm,
                                 const float* __restrict__ Bf, int CB, int B_pm,
                                 int Kp, float* __restrict__ X)
{
    int Ctot = 3 + CA + CB;
    size_t total = (size_t)NB * NPTS * Kp;
    size_t t = (size_t)blockIdx.x * blockDim.x + threadIdx.x;
    if (t >= total) return;
    int c = (int)(t % Kp);
    size_t bn = t / Kp;
    int n = (int)(bn % NPTS);
    int b = (int)(bn / NPTS);
    X[t] = (c < Ctot) ? read_src(loc, A, CA, A_pm, Bf, CB, B_pm, b, c, n) : 0.0f;
}

// ---------------------------------------------------------------------------
// H[b,n,c] = concat([loc;A;Bf])[c];  AGG[b,n,c] = sum over 3 NN j of H[b,j,c]
// Both point-major with zeroed pad columns.
// ---------------------------------------------------------------------------
__global__ void gather_agg_kernel(const float* __restrict__ loc,
                                  const float* __restrict__ A, int CA, int A_pm,
                                  const float* __restrict__ Bf, int CB, int B_pm,
                                  const int* __restrict__ idx, int Kp,
                                  float* __restrict__ H, float* __restrict__ AGG)
{
    int Ctot = 3 + CA + CB;
    size_t total = (size_t)NB * NPTS * Kp;
    size_t t = (size_t)blockIdx.x * blockDim.x + threadIdx.x;
    if (t >= total) return;
    int c = (int)(t % Kp);
    size_t bn = t / Kp;
    int n = (int)(bn % NPTS);
    int b = (int)(bn / NPTS);
    if (c >= Ctot) { H[t] = 0.0f; AGG[t] = 0.0f; return; }
    const int* ip = idx + ((size_t)b * NPTS + n) * 3;
    int j0 = ip[0], j1 = ip[1], j2 = ip[2];
    H[t]   = read_src(loc, A, CA, A_pm, Bf, CB, B_pm, b, c, n);
    AGG[t] = read_src(loc, A, CA, A_pm, Bf, CB, B_pm, b, c, j0)
           + read_src(loc, A, CA, A_pm, Bf, CB, B_pm, b, c, j1)
           + read_src(loc, A, CA, A_pm, Bf, CB, B_pm, b, c, j2);
}

// ---------------------------------------------------------------------------
// Bilinear grid sample (border clamp) of 3 maps -> pooled (B, NPTS, 448) pm.
// grid is point-major (B, NPTS, 2).
// ---------------------------------------------------------------------------
__global__ void grid_sample_kernel(const float* __restrict__ p0,
                                   const float* __restrict__ p1,
                                   const float* __restrict__ p2,
                                   const float* __restrict__ grid,
                                   float* __restrict__ pooled)
{
    const int CT = 448;
    size_t total = (size_t)NB * NPTS * CT;
    size_t t = (size_t)blockIdx.x * blockDim.x + threadIdx.x;
    if (t >= total) return;
    int c = (int)(t % CT);
    size_t bn = t / CT;
    int n = (int)(bn % NPTS);
    int b = (int)(bn / NPTS);

    float gx = grid[bn * 2 + 0];
    float gy = grid[bn * 2 + 1];

    const float* img; int H, W;
    if (c < 64)       { img = p0 + ((size_t)b * 64  +  c       ) * 112 * 112; H = 112; W = 112; }
    else if (c < 192) { img = p1 + ((size_t)b * 128 + (c - 64 )) * 56  * 56;  H = 56;  W = 56;  }
    else              { img = p2 + ((size_t)b * 256 + (c - 192)) * 28  * 28;  H = 28;  W = 28;  }

    float ix = ((gx + 1.0f) * (float)W - 1.0f) * 0.5f;
    float iy = ((gy + 1.0f) * (float)H - 1.0f) * 0.5f;
    float fx = floorf(ix), fy = floorf(iy);
    float wx = ix - fx,    wy = iy - fy;
    int x0 = clampi((int)fx,     0, W - 1);
    int x1 = clampi((int)fx + 1, 0, W - 1);
    int y0 = clampi((int)fy,     0, H - 1);
    int y1 = clampi((int)fy + 1, 0, H - 1);

    float v00 = img[(size_t)y0 * W + x0];
    float v01 = img[(size_t)y0 * W + x1];
    float v10 = img[(size_t)y1 * W + x0];
    float v11 = img[(size_t)y1 * W + x1];
    pooled[t] = v00 * (1.0f - wx) * (1.0f - wy) + v01 * wx * (1.0f - wy)
              + v10 * (1.0f - wx) * wy          + v11 * wx * wy;
}

// ---------------------------------------------------------------------------
// Brute-force 3-NN (excluding self) in 3D; batch's 4096 points staged in LDS
// (48 KB of the 320 KB WGP LDS).  idx: (B, NPTS, 3) int32.
// ---------------------------------------------------------------------------
__global__ void knn_kernel(const float* __restrict__ loc, int* __restrict__ idx)
{
    __shared__ float lx[NPTS];
    __shared__ float ly[NPTS];
    __shared__ float lz[NPTS];
    const int blocks_per_b = NPTS / 256;
    int b     = blockIdx.x / blocks_per_b;
    int pbase = (blockIdx.x % blocks_per_b) * 256;
    const float* L = loc + (size_t)b * 3 * NPTS;
    for (int i = threadIdx.x; i < NPTS; i += 256) {
        lx[i] = L[i];
        ly[i] = L[NPTS + i];
        lz[i] = L[2 * NPTS + i];
    }
    __syncthreads();

    int p = pbase + threadIdx.x;
    float px = lx[p], py = ly[p], pz = lz[p];
    float d0 = 3.4e38f, d1 = 3.4e38f, d2 = 3.4e38f;
    int   i0 = 0,       i1 = 0,       i2 = 0;
    for (int j = 0; j < NPTS; ++j) {
        if (j == p) continue;
        float dx = lx[j] - px, dy = ly[j] - py, dz = lz[j] - pz;
        float d = dx * dx + dy * dy + dz * dz;
        if (d < d0)      { d2 = d1; i2 = i1; d1 = d0; i1 = i0; d0 = d; i0 = j; }
        else if (d < d1) { d2 = d1; i2 = i1; d1 = d;  i1 = j; }
        else if (d < d2) { d2 = d;  i2 = j; }
    }
    int* ip = idx + ((size_t)b * NPTS + p) * 3;
    ip[0] = i0; ip[1] = i1; ip[2] = i2;
}

// ---------------------------------------------------------------------------
// out[0 .. B*3*N)        = x_loc + tanh(logits)      (channel-major)
// out[B*3*N .. +B*128*N) = feat                      (channel-major)
// logits (B,NPTS,3) pm, feat (B,NPTS,128) pm.
// ---------------------------------------------------------------------------
__global__ void finalize_kernel(const float* __restrict__ loc,
                                const float* __restrict__ logits,
                                const float* __restrict__ feat,
                                float* __restrict__ out)
{
    size_t total = (size_t)NB * 131 * NPTS;
    size_t t = (size_t)blockIdx.x * blockDim.x + threadIdx.x;
    if (t >= total) return;
    int n = (int)(t % NPTS);
    size_t bc = t / NPTS;
    int c = (int)(bc % 131);
    int b = (int)(bc / 131);
    if (c < 3) {
        size_t o = ((size_t)b * 3 + c) * NPTS + n;
        out[o] = loc[o] + tanhf(logits[((size_t)b * NPTS + n) * 3 + c]);
    } else {
        int cf = c - 3;
        out[(size_t)NB * 3 * NPTS + ((size_t)b * 128 + cf) * NPTS + n] =
            feat[((size_t)b * NPTS + n) * 128 + cf];
    }
}

// ---------------------------------------------------------------------------
extern "C" void kernel_launch(void* const* d_in, const int* in_sizes, int n_in,
                              void* d_out, int out_size, void* d_ws, size_t ws_size,
                              hipStream_t stream)
{
    const float* x_loc  = (const float*)d_in[0];
    const float* x_feat = (const float*)d_in[1];
    const float* p0     = (const float*)d_in[2];
    const float* p1     = (const float*)d_in[3];
    const float* p2     = (const float*)d_in[4];
    const float* Ws1 = (const float*)d_in[5];  const float* bs1 = (const float*)d_in[6];
    const float* Ws2 = (const float*)d_in[7];  const float* bs2 = (const float*)d_in[8];
    const float* Wg  = (const float*)d_in[9];  const float* bg  = (const float*)d_in[10];
    const float* Wl1 = (const float*)d_in[11]; const float* bl1 = (const float*)d_in[12];
    const float* Wl2 = (const float*)d_in[13]; const float* bl2 = (const float*)d_in[14];
    const float* Wr1 = (const float*)d_in[15]; const float* Wn1 = (const float*)d_in[16];
    const float* b1  = (const float*)d_in[17];
    const float* Wr2 = (const float*)d_in[18]; const float* Wn2 = (const float*)d_in[19];
    const float* b2  = (const float*)d_in[20];
    const float* Wr3 = (const float*)d_in[21]; const float* Wn3 = (const float*)d_in[22];
    const float* b3  = (const float*)d_in[23];
    const float* Wloc = (const float*)d_in[24]; const float* bloc = (const float*)d_in[25];

    float* out = (float*)d_out;
    float* ws  = (float*)d_ws;

    const size_t NBP = (size_t)NB * NPTS;
    // Activation workspace (point-major, padded K), phase-reused:
    float* P    = ws;               // 448 f/pt : X0(132) -> Xg(68) -> pooled(448) -> H(388/260) -> Xf(132)
    float* AGG  = P   + 448 * NBP;  // 388 f/pt
    float* F1   = AGG + 388 * NBP;  // 256 f/pt : H1+H2 early; FEAT1 later
    float* F2   = F1  + 256 * NBP;  // 256 f/pt : f (Wl2 out); FEAT2 later
    float* F3   = F2  + 256 * NBP;  // 128 f/pt : FEAT3
    float* G    = F3  + 128 * NBP;  //   2 f/pt : grid
    float* LOG  = G   +   2 * NBP;  //   3 f/pt : logits
    int*   IDX  = (int*)(LOG + 3 * NBP); // 3 ints/pt
    float* H1   = F1;               //  64 f/pt
    float* H2   = F1  + 64 * NBP;   //  64 f/pt
    // Padded weights:
    float* Ws1p  = (float*)(IDX + 3 * NBP);
    float* Wgp   = Ws1p  + 64  * 132;
    float* Wr1p  = Wgp   + 16  * 68;
    float* Wn1p  = Wr1p  + 256 * 388;
    float* Wr2p  = Wn1p  + 256 * 388;
    float* Wn2p  = Wr2p  + 256 * 260;
    float* Wr3p  = Wn2p  + 256 * 260;
    float* Wn3p  = Wr3p  + 128 * 260;
    float* Wlocp = Wn3p  + 128 * 260;

    auto blocks_for = [](size_t total) -> unsigned {
        return (unsigned)((total + 255) / 256);
    };
    auto pad_w = [&](const float* src, int M, int K, float* dst, int Mp, int Kp) {
        pad_w_kernel<<<blocks_for((size_t)Mp * Kp), 256, 0, stream>>>(src, M, K, dst, Mp, Kp);
    };
    auto gemm = [&](const float* W1c, const float* X1c, int K1p,
                    const float* W2c, const float* X2c, int K2p,
                    const float* biasc, float* o, int M, int Mstride, int dorelu) {
        int tiles  = ((M + 15) / 16) * (NPTS / 16) * NB;
        int blocks = (tiles + 7) / 8;
        gemm_wmma_kernel<<<blocks, 256, 0, stream>>>(W1c, X1c, K1p, W2c, X2c, K2p,
                                                     biasc, o, M, Mstride, dorelu, tiles);
    };

    // --- weight padding (tiny) -------------------------------------------
    pad_w(Ws1,  64, 131, Ws1p,  64, 132);
    pad_w(Wg,    2,  67, Wgp,   16,  68);
    pad_w(Wr1, 256, 387, Wr1p, 256, 388);
    pad_w(Wn1, 256, 387, Wn1p, 256, 388);
    pad_w(Wr2, 256, 259, Wr2p, 256, 260);
    pad_w(Wn2, 256, 259, Wn2p, 256, 260);
    pad_w(Wr3, 128, 259, Wr3p, 128, 260);
    pad_w(Wn3, 128, 259, Wn3p, 128, 260);
    pad_w(Wloc,  3, 131, Wlocp, 16, 132);

    // --- pipeline ---------------------------------------------------------
    // 1) X0 = [loc; x_feat] pm, Kp=132
    concat_pm_kernel<<<blocks_for(132 * NBP), 256, 0, stream>>>(
        x_loc, x_feat, 128, 0, nullptr, 0, 0, 132, P);
    // 2) H1 = relu(Ws1 X0 + bs1)
    gemm(Ws1p, P, 132, nullptr, nullptr, 0, bs1, H1, 64, 64, 1);
    // 3) H2 = relu(Ws2 H1 + bs2)   (Ws2 64x64 unpadded, direct)
    gemm(Ws2, H1, 64, nullptr, nullptr, 0, bs2, H2, 64, 64, 1);
    // 4) Xg = [loc; H2] pm, Kp=68
    concat_pm_kernel<<<blocks_for(68 * NBP), 256, 0, stream>>>(
        x_loc, H2, 64, 1, nullptr, 0, 0, 68, P);
    // 5) grid = Wg Xg + bg  (pm, stride 2)
    gemm(Wgp, P, 68, nullptr, nullptr, 0, bg, G, 2, 2, 0);
    // 6) pooled = bilinear sample of 3 maps (pm, 448)
    grid_sample_kernel<<<blocks_for(448 * NBP), 256, 0, stream>>>(p0, p1, p2, G, P);
    // 7) F1 = relu(Wl1 pooled + bl1)   (Wl1 256x448 direct)
    gemm(Wl1, P, 448, nullptr, nullptr, 0, bl1, F1, 256, 256, 1);
    // 8) f = Wl2 F1 + bl2 (no relu)    (Wl2 256x256 direct)
    gemm(Wl2, F1, 256, nullptr, nullptr, 0, bl2, F2, 256, 256, 0);
    // 9) KNN over x_loc (indices reused by all 3 pcd layers)
    knn_kernel<<<NB * (NPTS / 256), 256, 0, stream>>>(x_loc, IDX);
    // 10) H/AGG layer 1 (387 ch: loc + x_feat(cm) + f(pm)), Kp=388
    gather_agg_kernel<<<blocks_for(388 * NBP), 256, 0, stream>>>(
        x_loc, x_feat, 128, 0, F2, 256, 1, IDX, 388, P, AGG);
    // 11) FEAT1 = relu(Wr1 H + Wn1 AGG + b1)
    gemm(Wr1p, P, 388, Wn1p, AGG, 388, b1, F1, 256, 256, 1);
    // 12) H/AGG layer 2 (259 ch), Kp=260
    gather_agg_kernel<<<blocks_for(260 * NBP), 256, 0, stream>>>(
        x_loc, F1, 256, 1, nullptr, 0, 0, IDX, 260, P, AGG);
    // 13) FEAT2 = relu(Wr2 H + Wn2 AGG + b2)
    gemm(Wr2p, P, 260, Wn2p, AGG, 260, b2, F2, 256, 256, 1);
    // 14) H/AGG layer 3 (259 ch), Kp=260
    gather_agg_kernel<<<blocks_for(260 * NBP), 256, 0, stream>>>(
        x_loc, F2, 256, 1, nullptr, 0, 0, IDX, 260, P, AGG);
    // 15) FEAT3 = relu(Wr3 H + Wn3 AGG + b3)
    gemm(Wr3p, P, 260, Wn3p, AGG, 260, b3, F3, 128, 128, 1);
    // 16) Xf = [loc; FEAT3] pm, Kp=132
    concat_pm_kernel<<<blocks_for(132 * NBP), 256, 0, stream>>>(
        x_loc, F3, 128, 1, nullptr, 0, 0, 132, P);
    // 17) logits = Wloc Xf + bloc (pm, stride 3)
    gemm(Wlocp, P, 132, nullptr, nullptr, 0, bloc, LOG, 3, 3, 0);
    // 18) out = [loc + tanh(logits); FEAT3]
    finalize_kernel<<<blocks_for(131 * NBP), 256, 0, stream>>>(x_loc, LOG, F3, out);
}